// No_sparsity_52828097740972
// MI455X (gfx1250) — compile-verified
//
#include <hip/hip_runtime.h>

// Pure data-movement op: out[(i*n+j)] = concat(emb[i], emb[j], w[j]).
// ~1.09 GB stored, 0 FLOPs -> store-bandwidth bound (~47us at 23.3 TB/s).
// Strategy: per-block LDS tile staging + gfx1250 async LDS->global b128 stores,
// double-buffered, all global stores 16B-aligned.

typedef float v4f __attribute__((ext_vector_type(4)));

#if defined(__gfx1250__) && __has_builtin(__builtin_amdgcn_global_store_async_from_lds_b128)
#define USE_ASYNC_LDS_STORE 1
#else
#define USE_ASYNC_LDS_STORE 0
#endif

#if USE_ASYNC_LDS_STORE
typedef int v4i __attribute__((ext_vector_type(4)));
typedef __attribute__((address_space(1))) v4i* global_v4i_ptr;
typedef __attribute__((address_space(3))) v4i* lds_v4i_ptr;
#endif

__device__ __forceinline__ void wait_async_zero() {
#if USE_ASYNC_LDS_STORE
#if __has_builtin(__builtin_amdgcn_s_wait_asynccnt)
  __builtin_amdgcn_s_wait_asynccnt(0);
#else
  asm volatile("s_wait_asynccnt 0" ::: "memory");
#endif
#endif
}

__global__ __launch_bounds__(256) void interact_concat_kernel(
    const float* __restrict__ emb,   // [n,32]
    const float* __restrict__ sw,    // [n]
    float* __restrict__ out,         // [n*n, 65]
    int n) {
  constexpr int F  = 32;             // feature width
  constexpr int L  = 2 * F + 1;      // 65 floats per output row
  constexpr int TR = 64;             // rows per tile
  constexpr int TILE_F  = TR * L;    // 4160 floats = 16640 B (multiple of 16)
  constexpr int TILE_V4 = TILE_F / 4;// 1040 float4 chunks

  __shared__ float lds[2 * TILE_F];  // 33,280 B double buffer

  const int tid  = threadIdx.x;
  const int lane = tid & 31;
  const int wave = tid >> 5;         // 8 wave32s
  const int i    = blockIdx.x;       // block row index

  // 'a' column for this lane is constant for the whole block: keep in a VGPR.
  const float a_val = emb[i * F + lane];

  float* const out_block = out + (size_t)i * (size_t)n * (size_t)L;

  auto compose = [&](int buf, int j0) {
    float* t = &lds[buf * TILE_F];
    // Each wave fills 8 rows; LDS addresses (65*r + lane) are conflict-free.
    for (int r = wave; r < TR; r += 8) {
      const int j = j0 + r;
      t[r * L + lane]     = a_val;                // a-part (register)
      t[r * L + F + lane] = emb[j * F + lane];    // b-part (coalesced, L2-hot)
    }
    if (tid < TR) t[tid * L + 2 * F] = sw[j0 + tid];  // weight column
  };

  compose(0, 0);
  __syncthreads();

  const int ntiles = n / TR;         // 32 for n=2048
  for (int tile = 0; tile < ntiles; ++tile) {
    const int cur = tile & 1;
    float* const src = &lds[cur * TILE_F];
    float* const dst = out_block + (size_t)tile * TILE_F;  // 16B-aligned

#if USE_ASYNC_LDS_STORE
    // Issue bulk async LDS->global b128 stores (ASYNCcnt-tracked), then
    // overlap composition of the next tile while the DMA drains.
    for (int c = tid; c < TILE_V4; c += 256) {
      __builtin_amdgcn_global_store_async_from_lds_b128(
          (global_v4i_ptr)(dst + 4 * c),
          (lds_v4i_ptr)(src + 4 * c),
          0, 0);
    }
#else
    for (int c = tid; c < TILE_V4; c += 256) {
      const v4f v = *(const v4f*)(src + 4 * c);
      __builtin_nontemporal_store(v, (v4f*)(dst + 4 * c));
    }
#endif

    if (tile + 1 < ntiles) compose(cur ^ 1, (tile + 1) * TR);

    // Async stores must finish reading LDS before buffer `cur` is reused.
    wait_async_zero();
    __syncthreads();
  }
}

extern "C" void kernel_launch(void* const* d_in, const int* in_sizes, int n_in,
                              void* d_out, int out_size, void* d_ws, size_t ws_size,
                              hipStream_t stream) {
  const float* emb = (const float*)d_in[0];  // [n, 32] fp32
  const float* sw  = (const float*)d_in[1];  // [n] fp32
  float* out = (float*)d_out;                // [n*n, 65] fp32
  const int n = in_sizes[1];                 // 2048

  interact_concat_kernel<<<n, 256, 0, stream>>>(emb, sw, out, n);
}